// PyramidRROIAlign_71923522338816
// MI455X (gfx1250) — compile-verified
//
#include <hip/hip_runtime.h>
#include <math.h>

#define POOLED 14
#define CELLS  196          // 14*14
#define CB     16           // channels per TDM stage
#define PMAX   22           // max patch side (level0 bound is 21)
#define CHAN   256

typedef unsigned int v4u __attribute__((ext_vector_type(4)));
typedef int          v8i __attribute__((ext_vector_type(8)));
typedef int          v4i __attribute__((ext_vector_type(4)));

__device__ __forceinline__ int rfl(unsigned x) {
    return __builtin_amdgcn_readfirstlane((int)x);
}

// Issue one TDM load: 3-D tile (x=pw, y=ph, z=cb channels) from global `gsrc`
// (already offset to tile start) into LDS byte offset `ldsoff`.
// D# layout per CDNA5 ISA ch.8 (CDNA5 ISA).
__device__ __forceinline__ void tdm_load(const float* gsrc, unsigned ldsoff,
                                         int pw, int ph, int cb,
                                         unsigned rowStride, unsigned chStride)
{
    unsigned long long ga = (unsigned long long)(uintptr_t)gsrc;

    v4u g0;
    g0.x = (unsigned)rfl(1u);                                   // count=1, user descriptor
    g0.y = (unsigned)rfl(ldsoff);                               // lds_addr (bytes)
    g0.z = (unsigned)rfl((unsigned)(ga & 0xFFFFFFFFull));       // global_addr[31:0]
    g0.w = (unsigned)rfl(((unsigned)((ga >> 32) & 0x1FFFFFFull))
                         | (2u << 30));                         // global_addr[56:32] | type=2

    unsigned dim0 = (unsigned)pw, dim1 = (unsigned)ph;
    v8i g1;
    g1.s0 = rfl(2u << 16);                                      // data_size=2 (4 bytes)
    g1.s1 = rfl((dim0 & 0xFFFFu) << 16);                        // tensor_dim0[15:0] (abar addr=0)
    g1.s2 = rfl((dim0 >> 16) | ((dim1 & 0xFFFFu) << 16));       // dim0[31:16] | dim1[15:0]
    g1.s3 = rfl((dim1 >> 16) | ((unsigned)pw << 16));           // dim1[31:16] | tile_dim0
    g1.s4 = rfl(((unsigned)ph & 0xFFFFu)
                | (((unsigned)cb & 0xFFFFu) << 16));            // tile_dim1 | tile_dim2
    g1.s5 = rfl(rowStride);                                     // tensor_dim0_stride[31:0]
    g1.s6 = rfl((chStride & 0xFFFFu) << 16);                    // stride0[47:32]=0 | stride1[15:0]
    g1.s7 = rfl(chStride >> 16);                                // stride1[47:16]

    v4i g2;
    g2.x = rfl((unsigned)cb);                                   // tensor_dim2 = cb (no OOB)
    g2.y = 0; g2.z = 0; g2.w = 0;
    v4i g3 = {0, 0, 0, 0};
    v8i g4 = {0, 0, 0, 0, 0, 0, 0, 0};                          // extra operand (clang-23 form)

    __builtin_amdgcn_tensor_load_to_lds(g0, g1, g2, g3, g4, /*cpol=*/0);
}

__global__ void __launch_bounds__(256)
rroi_align_kernel(const float* __restrict__ x0, const float* __restrict__ x1,
                  const float* __restrict__ x2, const float* __restrict__ x3,
                  const float* __restrict__ b0, const float* __restrict__ b1,
                  const float* __restrict__ b2, const float* __restrict__ b3,
                  float* __restrict__ out)
{
    __shared__ float s_w00[CELLS], s_w01[CELLS], s_w10[CELLS], s_w11[CELLS];
    __shared__ int   s_off[CELLS];
    __shared__ int   s_minx, s_maxx, s_miny, s_maxy;
    extern __shared__ float dynbuf[];            // 2 * CB*PMAX*PMAX floats

    const int tid = threadIdx.x;
    const int gb  = blockIdx.x;                  // 0..2047
    const int L   = gb >> 9;
    const int n   = gb & 511;

    const float  scales[4] = {1.0f, 0.5f, 0.25f, 0.125f};
    const int    HS[4] = {200, 100, 50, 25};
    const int    WS[4] = {176,  88, 44, 22};
    const float* feats[4] = {x0, x1, x2, x3};
    const float* boxes[4] = {b0, b1, b2, b3};

    const int H = HS[L], W = WS[L];
    const float* feat = feats[L];
    const float* box  = boxes[L] + n * 7;

    if (tid == 0) { s_minx = 1 << 30; s_miny = 1 << 30; s_maxx = -1; s_maxy = -1; }
    __syncthreads();

    int xb = 0, yb = 0;
    if (tid < CELLS) {
        const float sc = scales[L];
        float cx = (box[0] * 2.4857954545454545f + 0.5f) * sc;   // 175/70.4
        float cy = ((box[1] + 40.0f) * 2.4875f + 0.5f) * sc;     // 199/80
        float rw = box[3] * 2.4857954545454545f * sc;
        float rh = box[4] * 2.4875f * sc;
        float th = box[6];

        int py = tid / POOLED, px = tid - py * POOLED;
        float fw = ((float)px + 0.5f) * (1.0f / POOLED) - 0.5f;
        float fh = ((float)py + 0.5f) * (1.0f / POOLED) - 0.5f;
        float lx = fw * rw, ly = fh * rh;
        float cth = cosf(th), sth = sinf(th);
        float xx = cx + lx * cth - ly * sth;
        float yy = cy + lx * sth + ly * cth;

        bool valid = (yy > -1.0f) && (yy < (float)H) && (xx > -1.0f) && (xx < (float)W);
        yy = fminf(fmaxf(yy, 0.0f), (float)(H - 1));
        xx = fminf(fmaxf(xx, 0.0f), (float)(W - 1));
        float y0f = floorf(yy), x0f = floorf(xx);
        int   iy0 = (int)y0f,   ix0 = (int)x0f;
        float lyf = yy - y0f,   lxf = xx - x0f;
        float hyf = 1.0f - lyf, hxf = 1.0f - lxf;

        // Map the (possibly clamped) taps onto an aligned pair base in [0, H-2]/[0, W-2].
        float wyA, wyB, wxA, wxB;
        if (iy0 >= H - 1) { yb = H - 2; wyA = 0.0f; wyB = 1.0f; }
        else              { yb = iy0;   wyA = hyf;  wyB = lyf;  }
        if (ix0 >= W - 1) { xb = W - 2; wxA = 0.0f; wxB = 1.0f; }
        else              { xb = ix0;   wxA = hxf;  wxB = lxf;  }

        float v = valid ? 1.0f : 0.0f;
        s_w00[tid] = v * wyA * wxA;  s_w01[tid] = v * wyA * wxB;
        s_w10[tid] = v * wyB * wxA;  s_w11[tid] = v * wyB * wxB;

        atomicMin(&s_minx, xb); atomicMax(&s_maxx, xb);
        atomicMin(&s_miny, yb); atomicMax(&s_maxy, yb);
    }
    __syncthreads();

    const int px0 = s_minx, py0 = s_miny;
    int pw = s_maxx - px0 + 2; if (pw > PMAX) pw = PMAX;
    int ph = s_maxy - py0 + 2; if (ph > PMAX) ph = PMAX;

    if (tid < CELLS) s_off[tid] = (yb - py0) * pw + (xb - px0);
    __syncthreads();

    const unsigned rowStride = (unsigned)W;          // elements
    const unsigned chStride  = (unsigned)(H * W);    // elements (< 2^16)
    const int      bufStride = CB * PMAX * PMAX;
    const float*   tileBase  = feat + (size_t)py0 * W + px0;

    const bool isWave0 = (tid < 32);                 // wave-uniform (wave32)

    if (isWave0) {
        tdm_load(tileBase, (unsigned)(uintptr_t)&dynbuf[0], pw, ph, CB,
                 rowStride, chStride);
    }

    float* outBoxBase = out + (size_t)gb * (CHAN * CELLS);
    const int NB = CHAN / CB;                        // 16 channel blocks

    for (int blk = 0; blk < NB; ++blk) {
        if (isWave0) __builtin_amdgcn_s_wait_tensorcnt(0);   // current buffer ready
        __syncthreads();

        if (blk + 1 < NB && isWave0) {               // prefetch next block (double buffer)
            const float* src  = tileBase + (size_t)(blk + 1) * CB * chStride;
            float*       dbuf = &dynbuf[((blk + 1) & 1) * bufStride];
            tdm_load(src, (unsigned)(uintptr_t)dbuf, pw, ph, CB,
                     rowStride, chStride);
        }

        const float* buf  = &dynbuf[(blk & 1) * bufStride];
        float*       outp = outBoxBase + blk * CB * CELLS;
        const int    plane = ph * pw;

        for (int i = tid; i < CB * CELLS; i += 256) {
            int cl   = i / CELLS;
            int cell = i - cl * CELLS;
            const float* base = buf + cl * plane + s_off[cell];
            float v00 = base[0],  v01 = base[1];
            float v10 = base[pw], v11 = base[pw + 1];
            outp[i] = s_w00[cell] * v00 + s_w01[cell] * v01
                    + s_w10[cell] * v10 + s_w11[cell] * v11;
        }
        __syncthreads();
    }
}

extern "C" void kernel_launch(void* const* d_in, const int* in_sizes, int n_in,
                              void* d_out, int out_size, void* d_ws, size_t ws_size,
                              hipStream_t stream)
{
    (void)in_sizes; (void)n_in; (void)d_ws; (void)ws_size; (void)out_size;
    const float* x0 = (const float*)d_in[0];
    const float* x1 = (const float*)d_in[1];
    const float* x2 = (const float*)d_in[2];
    const float* x3 = (const float*)d_in[3];
    const float* b0 = (const float*)d_in[4];
    const float* b1 = (const float*)d_in[5];
    const float* b2 = (const float*)d_in[6];
    const float* b3 = (const float*)d_in[7];

    size_t shmem = (size_t)2 * CB * PMAX * PMAX * sizeof(float);   // 61,952 B
    rroi_align_kernel<<<dim3(2048), dim3(256), shmem, stream>>>(
        x0, x1, x2, x3, b0, b1, b2, b3, (float*)d_out);
}